// ExponentialMovingAverage_90280212562197
// MI455X (gfx1250) — compile-verified
//
#include <hip/hip_runtime.h>

typedef __attribute__((ext_vector_type(2))) float v2f;
typedef __attribute__((ext_vector_type(4))) float v4f;
typedef __attribute__((ext_vector_type(8))) float v8f;

namespace {
constexpr int kT = 8192;           // time steps per chain
constexpr int kC = 128;            // channels
constexpr int kB = 64;             // batch
constexpr int kChunk = 256;        // 16 positions x 16 columns per wave-iteration
constexpr int kWavesPerBlock = 4;
constexpr int kBlockSize = kWavesPerBlock * 32;   // wave32
}

__global__ __launch_bounds__(kBlockSize)
void ema_scan_wmma_f32(const float* __restrict__ x,
                       const float* __restrict__ weights,
                       float* __restrict__ out)
{
    const int lane  = threadIdx.x & 31;
    const int wave  = threadIdx.x >> 5;
    const int chain = blockIdx.x * kWavesPerBlock + wave;   // chain = b*kC + c
    const int c     = chain & (kC - 1);
    const long long base = (long long)chain * kT;

    float w = weights[c];
    w = fminf(fmaxf(w, 0.0f), 1.0f);
    const float a = 1.0f - w;

    // power ladder of a
    const float a2  = a * a;
    const float a4  = a2 * a2;
    const float a8  = a4 * a4;
    const float r   = a8 * a8;     // a^16  (chunk-to-chunk ratio)
    const float r2  = r * r;
    const float r4  = r2 * r2;
    const float r8  = r4 * r4;
    const float r16 = r8 * r8;     // a^256 (iteration-to-iteration ratio)
    const float a9  = a8 * a;

    const int  n  = lane & 15;     // column index (chunk id within group of 16)
    const bool hi = lane >= 16;

    // ---- Build A matrix: L'[m][k] = (m >= k) ? w * a^(m-k) : 0 ----
    // 16x4 f32 A layout: lanes 0-15 -> M=lane, K=4s+{0,1}; lanes 16-31 -> M=lane-16, K=4s+{2,3}
    v2f Amat[4];
    {
        const int m = n;
        #pragma unroll
        for (int s = 0; s < 4; ++s) {
            #pragma unroll
            for (int j = 0; j < 2; ++j) {
                const int k = 4 * s + (hi ? 2 : 0) + j;
                const int d = m - k;
                float p = 1.0f;
                p *= (d & 1) ? a  : 1.0f;
                p *= (d & 2) ? a2 : 1.0f;
                p *= (d & 4) ? a4 : 1.0f;
                p *= (d & 8) ? a8 : 1.0f;
                Amat[s][j] = (d >= 0) ? (w * p) : 0.0f;
            }
        }
    }

    // per-lane r^n for carry injection into column n
    float rn = 1.0f;
    rn *= (n & 1) ? r  : 1.0f;
    rn *= (n & 2) ? r2 : 1.0f;
    rn *= (n & 4) ? r4 : 1.0f;
    rn *= (n & 8) ? r8 : 1.0f;

    // output coefficient base: a^(mbase+1), mbase = hi ? 8 : 0
    const float coef0 = hi ? a9 : a;

    const int boffB   = 16 * n + (hi ? 2 : 0);   // B-matrix load offset (float2, stride 4 per slice)
    const int boffOut = 16 * n + (hi ? 8 : 0);   // contiguous 8-float store offset

    // y_{-1} = x_0 makes y_0 == x_0, matching the reference's init handling
    float carry = x[base];

    for (int t0 = 0; t0 < kT; t0 += kChunk) {
        const float* px = x + base + t0;

        // ---- Load B: 4x16 slices, column n = chunk, row k = position in chunk ----
        v2f Bm[4];
        #pragma unroll
        for (int s = 0; s < 4; ++s)
            Bm[s] = *(const v2f*)(px + boffB + 4 * s);

        // ---- S = L' x B : 16 chunk-local inclusive scans via 4 chained f32 WMMAs ----
        v8f S = {0.0f, 0.0f, 0.0f, 0.0f, 0.0f, 0.0f, 0.0f, 0.0f};
        #pragma unroll
        for (int s = 0; s < 4; ++s)
            S = __builtin_amdgcn_wmma_f32_16x16x4_f32(false, Amat[s], false, Bm[s],
                                                      (short)0, S, false, false);

        // ---- Cross-column carry scan (ratio r = a^16) ----
        // S[15][n] sits in lanes 16..31, component 7
        float s15 = __shfl(S[7], 16 + n, 32);

        float p  = s15;   // inclusive: I_n = sum_{j<=n} r^(n-j) * s15_j
        float rp = r;
        #pragma unroll
        for (int o = 1; o < 16; o <<= 1) {
            float q = __shfl_up(p, o, 32);
            p = (n >= o) ? fmaf(rp, q, p) : p;
            rp *= rp;
        }
        const float I15 = __shfl(p, 15, 32);
        float E = __shfl_up(p, 1, 32);         // exclusive: E_n = I_{n-1}, E_0 = 0
        E = (n >= 1) ? E : 0.0f;
        const float cn = fmaf(rn, carry, E);   // carry entering column n

        // ---- y[m][n] = a^(m+1) * c_n + S[m][n]; 8 contiguous outputs per lane ----
        float yv[8];
        float cf = coef0;
        #pragma unroll
        for (int i = 0; i < 8; ++i) {
            yv[i] = fmaf(cf, cn, S[i]);
            cf *= a;
        }
        float* po = out + base + t0 + boffOut;
        v4f lo = {yv[0], yv[1], yv[2], yv[3]};
        v4f hi4 = {yv[4], yv[5], yv[6], yv[7]};
        *(v4f*)(po)     = lo;
        *(v4f*)(po + 4) = hi4;

        carry = fmaf(r16, carry, I15);
    }
}

extern "C" void kernel_launch(void* const* d_in, const int* in_sizes, int n_in,
                              void* d_out, int out_size, void* d_ws, size_t ws_size,
                              hipStream_t stream)
{
    const float* x       = (const float*)d_in[0];
    const float* weights = (const float*)d_in[1];
    float* out           = (float*)d_out;

    const int chains = kB * kC;                    // 8192 independent scan chains
    dim3 grid(chains / kWavesPerBlock);            // 2048 blocks
    dim3 block(kBlockSize);                        // 4 waves of 32
    ema_scan_wmma_f32<<<grid, block, 0, stream>>>(x, weights, out);
}